// AutoregU_83494164234418
// MI455X (gfx1250) — compile-verified
//
#include <hip/hip_runtime.h>
#include <math.h>
#include <stdint.h>

typedef __attribute__((ext_vector_type(2))) float v2f;
typedef __attribute__((ext_vector_type(8))) float v8f;

#define T_STEPS 6
#define IN_F    11
#define XPAD    12      // padded K for the 11-wide terms (multiple of 4)
#define HDIM    128
#define OUT_F   3

// ---------------------------------------------------------------------------
// WMMA: D = A(16x4,f32) * B(4x16,f32) + C(16x16,f32)   -- exact fp32 path
// ---------------------------------------------------------------------------
__device__ __forceinline__ v8f wmma4(v2f a, v2f b, v8f c) {
  return __builtin_amdgcn_wmma_f32_16x16x4_f32(
      /*neg_a=*/false, a, /*neg_b=*/false, b,
      /*c_mod=*/(short)0, c, /*reuse_a=*/false, /*reuse_b=*/false);
}

// Packed-B layout: P[(k/4)*HDIM*4 + col*4 + (k%4)] = (k < Kv) ? W[k][col] : 0
// -> lane (lo,hi) fetches its (b.x,b.y) pair as ONE aligned 8-byte load.
template <int KP, bool HASW1>
__device__ __forceinline__ void gemm_term(const float* sA, int ldA,
                                          const float* __restrict__ P0,
                                          const float* __restrict__ P1,
                                          int col, int lo, int hi,
                                          v8f& acc0, v8f& acc1) {
  #pragma unroll
  for (int k0 = 0; k0 < KP; k0 += 4) {
    const int ka = k0 + hi * 2;
    v2f a = *(const v2f*)&sA[lo * ldA + ka];               // ds_load_b64
    const size_t pb = (size_t)(k0 >> 2) * (HDIM * 4) + col * 4 + hi * 2;
    v2f b0 = *(const v2f*)&P0[pb];                         // global_load_b64
    acc0 = wmma4(a, b0, acc0);
    if (HASW1) {
      v2f b1 = *(const v2f*)&P1[pb];
      acc1 = wmma4(a, b1, acc1);
    }
  }
}

// ---------------------------------------------------------------------------
// Async global -> LDS staging (CDNA5 ASYNCcnt path).
// lds_byte = per-lane LDS byte address (low 32 bits of generic pointer).
// ---------------------------------------------------------------------------
__device__ __forceinline__ void async_ld_b128(unsigned lds_byte, const float* g) {
  asm volatile("global_load_async_to_lds_b128 %0, %1, off"
               :: "v"(lds_byte), "v"(g) : "memory");
}
__device__ __forceinline__ void wait_async0() {
  asm volatile("s_wait_asynccnt 0" ::: "memory");
}
__device__ __forceinline__ unsigned lds_base_of(const void* p) {
  return (unsigned)(uintptr_t)p;   // generic LDS ptr: addr[31:0] == LDS address
}

// ---------------------------------------------------------------------------
// Utility kernels
// ---------------------------------------------------------------------------
__global__ void zero_kernel(float* __restrict__ p, long n) {
  long i = (long)blockIdx.x * blockDim.x + threadIdx.x;
  if (i < n) p[i] = 0.f;
}

// Repack W[Kv x HDIM] (row-major) into padded WMMA-B layout, KP rows.
__global__ void repack_kernel(const float* __restrict__ W, float* __restrict__ P,
                              int Kv, int Kp) {
  int i = blockIdx.x * blockDim.x + threadIdx.x;
  if (i >= Kp * HDIM) return;
  int k = i / HDIM, c = i - k * HDIM;
  float v = (k < Kv) ? W[(size_t)k * HDIM + c] : 0.f;
  P[((size_t)(k >> 2) * HDIM + c) * 4 + (k & 3)] = v;
}

__global__ void deg_kernel(const int* __restrict__ src, const int* __restrict__ dst,
                           float* __restrict__ deg, int E) {
  int e = blockIdx.x * blockDim.x + threadIdx.x;
  if (e >= E) return;
  int s = src[e], d = dst[e];
  if (s != d) atomicAdd(&deg[s], 1.f);
}

__global__ void norm_kernel(const int* __restrict__ src, const int* __restrict__ dst,
                            const float* __restrict__ deg, float* __restrict__ nrm, int E) {
  int e = blockIdx.x * blockDim.x + threadIdx.x;
  if (e >= E) return;
  int s = src[e], d = dst[e];
  float w   = (s != d) ? 1.f : 0.f;
  float ds_ = deg[s], dd = deg[d];
  float dis = (ds_ > 0.f) ? rsqrtf(fmaxf(ds_, 1.f)) : 0.f;
  float did = (dd  > 0.f) ? rsqrtf(fmaxf(dd,  1.f)) : 0.f;
  nrm[e] = -dis * w * did;
}

__global__ void xm_kernel(const float* __restrict__ Xseq,
                          const float* __restrict__ uprev, const float* __restrict__ vprev,
                          float* __restrict__ xm, int N, int t) {
  int n = blockIdx.x * blockDim.x + threadIdx.x;
  if (n >= N) return;
  const float* x = Xseq + ((size_t)t * N + n) * IN_F;
  float* o = xm + (size_t)n * XPAD;
  o[0] = x[0]; o[1] = x[1]; o[2] = x[2];
  if (t == 0) { o[3] = x[3]; o[4] = x[4]; o[5] = x[5]; }
  else        { o[3] = uprev[n*3+0]; o[4] = uprev[n*3+1]; o[5] = uprev[n*3+2]; }
  o[6] = x[6]; o[7] = x[7];
  if (t == 0) { o[8] = x[8]; o[9] = x[9]; o[10] = x[10]; }
  else        { o[8] = vprev[n*3+0]; o[9] = vprev[n*3+1]; o[10] = vprev[n*3+2]; }
  o[11] = 0.f;  // pad column (zero contribution in GEMM)
}

__global__ void scatter_x_kernel(const int* __restrict__ src, const int* __restrict__ dst,
                                 const float* __restrict__ nrm, const float* __restrict__ xm,
                                 float* __restrict__ agg, int E) {
  int e = blockIdx.x * blockDim.x + threadIdx.x;
  if (e >= E) return;
  float w = nrm[e];
  if (w == 0.f) return;
  int s = src[e], d = dst[e];
  const float* xp = xm + (size_t)s * XPAD;
  float* op = agg + (size_t)d * XPAD;
  #pragma unroll
  for (int c = 0; c < IN_F; ++c) atomicAdd(&op[c], w * xp[c]);
}

__global__ void scatter_h_kernel(const int* __restrict__ src, const int* __restrict__ dst,
                                 const float* __restrict__ nrm, const float* __restrict__ v,
                                 float* __restrict__ agg, int E) {
  long tid = (long)blockIdx.x * blockDim.x + threadIdx.x;
  int e = (int)(tid >> 4);
  int part = (int)(tid & 15);
  if (e >= E) return;
  float w = nrm[e];
  if (w == 0.f) return;
  int s = src[e], d = dst[e];
  const float* xp = v + (size_t)s * HDIM + part * 8;
  float* op = agg + (size_t)d * HDIM + part * 8;
  #pragma unroll
  for (int j = 0; j < 8; ++j) atomicAdd(&op[j], w * xp[j]);
}

// ---------------------------------------------------------------------------
// Shared staging: 4 tiles {A0[16xXPAD], A1[16xXPAD], B0[16xHDIM], B1[16xHDIM]}
// All tiles are contiguous slabs in global memory -> async b128 copies.
// ---------------------------------------------------------------------------
#define OFF_X0  0
#define OFF_X1  (16 * XPAD)
#define OFF_H0  (2 * 16 * XPAD)
#define OFF_H1  (2 * 16 * XPAD + 16 * HDIM)
#define SMEM_FLTS (2 * 16 * XPAD + 2 * 16 * HDIM)

__device__ __forceinline__ void stage_tiles(float* smem, int row0, int N,
                                            const float* gx0, const float* gx1,
                                            const float* gh0, const float* gh1,
                                            int tidx) {
  if (row0 + 16 <= N) {
    const unsigned base = lds_base_of(smem);
    const float* px0 = gx0 + (size_t)row0 * XPAD;
    const float* px1 = gx1 + (size_t)row0 * XPAD;
    const float* ph0 = gh0 + (size_t)row0 * HDIM;
    const float* ph1 = gh1 + (size_t)row0 * HDIM;
    if (tidx < 48) {  // 16*12 floats = 48 x 16B
      async_ld_b128(base + (OFF_X0 + tidx * 4) * 4, px0 + tidx * 4);
      async_ld_b128(base + (OFF_X1 + tidx * 4) * 4, px1 + tidx * 4);
    }
    #pragma unroll
    for (int j = 0; j < 2; ++j) {  // 16*128 floats = 512 x 16B, 256 thr
      int idx = j * 256 + tidx;
      async_ld_b128(base + (OFF_H0 + idx * 4) * 4, ph0 + idx * 4);
      async_ld_b128(base + (OFF_H1 + idx * 4) * 4, ph1 + idx * 4);
    }
    wait_async0();
  } else {  // guarded scalar tail (never taken when N % 16 == 0)
    for (int i = tidx; i < 16 * XPAD; i += 256) {
      int r = i / XPAD, c = i - r * XPAD;
      int g = row0 + r;  bool ok = (g < N);
      smem[OFF_X0 + i] = ok ? gx0[(size_t)g * XPAD + c] : 0.f;
      smem[OFF_X1 + i] = ok ? gx1[(size_t)g * XPAD + c] : 0.f;
    }
    for (int i = tidx; i < 16 * HDIM; i += 256) {
      int r = i >> 7, c = i & (HDIM - 1);
      int g = row0 + r;  bool ok = (g < N);
      smem[OFF_H0 + i] = ok ? gh0[(size_t)g * HDIM + c] : 0.f;
      smem[OFF_H1 + i] = ok ? gh1[(size_t)g * HDIM + c] : 0.f;
    }
  }
  __syncthreads();
}

// ---------------------------------------------------------------------------
// Gates kernel: z,r share the A operand -> paired WMMA chains.
// ---------------------------------------------------------------------------
__global__ __launch_bounds__(256) void gates_kernel(
    const float* __restrict__ xm,  const float* __restrict__ aggx,
    const float* __restrict__ h,   const float* __restrict__ aggh,
    const float* __restrict__ pxz0, const float* __restrict__ pxz1,
    const float* __restrict__ pxr0, const float* __restrict__ pxr1,
    const float* __restrict__ phz0, const float* __restrict__ phz1,
    const float* __restrict__ phr0, const float* __restrict__ phr1,
    const float* __restrict__ xzb, const float* __restrict__ xrb,
    const float* __restrict__ hzb, const float* __restrict__ hrb,
    float* __restrict__ zbuf, float* __restrict__ hrbuf, int N) {
  __shared__ __align__(16) float smem[SMEM_FLTS];
  const int row0 = blockIdx.x * 16;
  __builtin_prefetch(phz0, 0, 3);
  __builtin_prefetch(phr0, 0, 3);

  stage_tiles(smem, row0, N, xm, aggx, h, aggh, threadIdx.x);

  const int lane = threadIdx.x & 31;
  const int lo = lane & 15, hi = lane >> 4;
  const int col = (threadIdx.x >> 5) * 16 + lo;   // global output column

  v8f az = {}; v8f ar = {};
  gemm_term<XPAD, true>(smem + OFF_X0, XPAD, pxz0, pxr0, col, lo, hi, az, ar);
  gemm_term<XPAD, true>(smem + OFF_X1, XPAD, pxz1, pxr1, col, lo, hi, az, ar);
  gemm_term<HDIM, true>(smem + OFF_H0, HDIM, phz0, phr0, col, lo, hi, az, ar);
  gemm_term<HDIM, true>(smem + OFF_H1, HDIM, phz1, phr1, col, lo, hi, az, ar);

  const float bz = xzb[col] + hzb[col];
  const float br = xrb[col] + hrb[col];
  #pragma unroll
  for (int i = 0; i < 8; ++i) {
    int m = i + hi * 8;          // C/D layout: VGPR i -> M=i / M=i+8
    int g = row0 + m;
    if (g < N) {
      float zz = 1.f / (1.f + expf(-(az[i] + bz)));
      float rr = 1.f / (1.f + expf(-(ar[i] + br)));
      float hv = smem[OFF_H0 + m * HDIM + col];
      zbuf[(size_t)g * HDIM + col]  = zz;
      hrbuf[(size_t)g * HDIM + col] = rr * hv;
    }
  }
}

// ---------------------------------------------------------------------------
// Candidate kernel: h_t = tanh(...); h = z*h + (1-z)*h_t (in place).
// ---------------------------------------------------------------------------
__global__ __launch_bounds__(256) void cand_kernel(
    const float* __restrict__ xm,  const float* __restrict__ aggx,
    const float* __restrict__ hrbuf, const float* __restrict__ agghr,
    const float* __restrict__ pxh0, const float* __restrict__ pxh1,
    const float* __restrict__ phh0, const float* __restrict__ phh1,
    const float* __restrict__ xhb, const float* __restrict__ hhb,
    const float* __restrict__ zbuf, float* __restrict__ h, int N) {
  __shared__ __align__(16) float smem[SMEM_FLTS];
  const int row0 = blockIdx.x * 16;
  __builtin_prefetch(phh0, 0, 3);

  stage_tiles(smem, row0, N, xm, aggx, hrbuf, agghr, threadIdx.x);

  const int lane = threadIdx.x & 31;
  const int lo = lane & 15, hi = lane >> 4;
  const int col = (threadIdx.x >> 5) * 16 + lo;

  v8f acc = {}; v8f dummy = {};
  gemm_term<XPAD, false>(smem + OFF_X0, XPAD, pxh0, nullptr, col, lo, hi, acc, dummy);
  gemm_term<XPAD, false>(smem + OFF_X1, XPAD, pxh1, nullptr, col, lo, hi, acc, dummy);
  gemm_term<HDIM, false>(smem + OFF_H0, HDIM, phh0, nullptr, col, lo, hi, acc, dummy);
  gemm_term<HDIM, false>(smem + OFF_H1, HDIM, phh1, nullptr, col, lo, hi, acc, dummy);

  const float bb = xhb[col] + hhb[col];
  #pragma unroll
  for (int i = 0; i < 8; ++i) {
    int m = i + hi * 8;
    int g = row0 + m;
    if (g < N) {
      float ht = tanhf(acc[i] + bb);
      size_t idx = (size_t)g * HDIM + col;
      float zz = zbuf[idx];
      float hv = h[idx];
      h[idx] = zz * hv + (1.f - zz) * ht;
    }
  }
}

// ---------------------------------------------------------------------------
// Head: u = h @ headW + headb; write outs[t]; update u_prev / v_prev.
// ---------------------------------------------------------------------------
__global__ void head_kernel(const float* __restrict__ h,
                            const float* __restrict__ headW, const float* __restrict__ headb,
                            const float* __restrict__ Xseq, const float* __restrict__ xm,
                            float* __restrict__ out, float* __restrict__ uprev,
                            float* __restrict__ vprev, int N, int t) {
  int n = blockIdx.x * blockDim.x + threadIdx.x;
  if (n >= N) return;
  float a0 = headb[0], a1 = headb[1], a2 = headb[2];
  const float* hp = h + (size_t)n * HDIM;
  #pragma unroll 4
  for (int c = 0; c < HDIM; ++c) {
    float hv = hp[c];
    a0 += hv * headW[c * OUT_F + 0];
    a1 += hv * headW[c * OUT_F + 1];
    a2 += hv * headW[c * OUT_F + 2];
  }
  size_t ob = ((size_t)t * N + n) * OUT_F;
  out[ob + 0] = a0; out[ob + 1] = a1; out[ob + 2] = a2;

  float dt = 1.f;
  if (t < T_STEPS - 1)
    dt = Xseq[((size_t)(t + 1) * N + n) * IN_F + 6] - Xseq[((size_t)t * N + n) * IN_F + 6];
  float u[3] = {a0, a1, a2};
  #pragma unroll
  for (int j = 0; j < 3; ++j) {
    uprev[n * 3 + j] = u[j];
    vprev[n * 3 + j] = (u[j] - xm[(size_t)n * XPAD + 3 + j]) / dt;
  }
}

// ---------------------------------------------------------------------------
// Host-side launch
// ---------------------------------------------------------------------------
static inline int cdiv(long a, long b) { return (int)((a + b - 1) / b); }

extern "C" void kernel_launch(void* const* d_in, const int* in_sizes, int n_in,
                              void* d_out, int out_size, void* d_ws, size_t ws_size,
                              hipStream_t stream) {
  const float* Xseq = (const float*)d_in[0];
  const int*   eidx = (const int*)d_in[1];
  const int N = in_sizes[0] / (T_STEPS * IN_F);
  const int E = in_sizes[1] / 2;
  const int* src = eidx;
  const int* dst = eidx + E;

  // Param order: insertion order (xz,hz,xr,hr,xh,hh,head) or tree-flatten
  // sorted order (head_W, head_b, hh, hr, hz, xh, xr, xz). Detect via size.
  const float *xzW,*xzb,*hzW,*hzb,*xrW,*xrb,*hrW,*hrb,*xhW,*xhb,*hhW,*hhb,*headW,*headb;
  if (in_sizes[2] == 2 * IN_F * HDIM) {
    xzW=(const float*)d_in[2];  xzb=(const float*)d_in[3];
    hzW=(const float*)d_in[4];  hzb=(const float*)d_in[5];
    xrW=(const float*)d_in[6];  xrb=(const float*)d_in[7];
    hrW=(const float*)d_in[8];  hrb=(const float*)d_in[9];
    xhW=(const float*)d_in[10]; xhb=(const float*)d_in[11];
    hhW=(const float*)d_in[12]; hhb=(const float*)d_in[13];
    headW=(const float*)d_in[14]; headb=(const float*)d_in[15];
  } else {
    headW=(const float*)d_in[2]; headb=(const float*)d_in[3];
    hhW=(const float*)d_in[4];  hhb=(const float*)d_in[5];
    hrW=(const float*)d_in[6];  hrb=(const float*)d_in[7];
    hzW=(const float*)d_in[8];  hzb=(const float*)d_in[9];
    xhW=(const float*)d_in[10]; xhb=(const float*)d_in[11];
    xrW=(const float*)d_in[12]; xrb=(const float*)d_in[13];
    xzW=(const float*)d_in[14]; xzb=(const float*)d_in[15];
  }
  float* out = (float*)d_out;

  // Workspace layout (all sub-buffer byte offsets are 16B-aligned)
  float* p = (float*)d_ws;
  float* deg   = p; p += N;
  float* nrm   = p; p += E;
  float* h     = p; p += (size_t)N * HDIM;
  float* xm    = p; p += (size_t)N * XPAD;
  float* aggx  = p; p += (size_t)N * XPAD;
  float* aggh  = p; p += (size_t)N * HDIM;
  float* agghr = p; p += (size_t)N * HDIM;
  float* zbuf  = p; p += (size_t)N * HDIM;
  float* hrbuf = p; p += (size_t)N * HDIM;
  float* uprev = p; p += (size_t)N * OUT_F;
  float* vprev = p; p += (size_t)N * OUT_F;
  // repacked weights
  float* pxz0 = p; p += XPAD * HDIM;
  float* pxz1 = p; p += XPAD * HDIM;
  float* pxr0 = p; p += XPAD * HDIM;
  float* pxr1 = p; p += XPAD * HDIM;
  float* pxh0 = p; p += XPAD * HDIM;
  float* pxh1 = p; p += XPAD * HDIM;
  float* phz0 = p; p += HDIM * HDIM;
  float* phz1 = p; p += HDIM * HDIM;
  float* phr0 = p; p += HDIM * HDIM;
  float* phr1 = p; p += HDIM * HDIM;
  float* phh0 = p; p += HDIM * HDIM;
  float* phh1 = p; p += HDIM * HDIM;

  const int TB = 256;

  // --- one-time (per launch) init ------------------------------------------
  zero_kernel<<<cdiv(N, TB), TB, 0, stream>>>(deg, N);
  zero_kernel<<<cdiv((long)N * HDIM, TB), TB, 0, stream>>>(h, (long)N * HDIM);
  zero_kernel<<<cdiv((long)N * OUT_F, TB), TB, 0, stream>>>(uprev, (long)N * OUT_F);
  zero_kernel<<<cdiv((long)N * OUT_F, TB), TB, 0, stream>>>(vprev, (long)N * OUT_F);
  deg_kernel<<<cdiv(E, TB), TB, 0, stream>>>(src, dst, deg, E);
  norm_kernel<<<cdiv(E, TB), TB, 0, stream>>>(src, dst, deg, nrm, E);

  const int RX = cdiv((long)XPAD * HDIM, TB), RH = cdiv((long)HDIM * HDIM, TB);
  repack_kernel<<<RX, TB, 0, stream>>>(xzW,                 pxz0, IN_F, XPAD);
  repack_kernel<<<RX, TB, 0, stream>>>(xzW + IN_F * HDIM,   pxz1, IN_F, XPAD);
  repack_kernel<<<RX, TB, 0, stream>>>(xrW,                 pxr0, IN_F, XPAD);
  repack_kernel<<<RX, TB, 0, stream>>>(xrW + IN_F * HDIM,   pxr1, IN_F, XPAD);
  repack_kernel<<<RX, TB, 0, stream>>>(xhW,                 pxh0, IN_F, XPAD);
  repack_kernel<<<RX, TB, 0, stream>>>(xhW + IN_F * HDIM,   pxh1, IN_F, XPAD);
  repack_kernel<<<RH, TB, 0, stream>>>(hzW,                 phz0, HDIM, HDIM);
  repack_kernel<<<RH, TB, 0, stream>>>(hzW + HDIM * HDIM,   phz1, HDIM, HDIM);
  repack_kernel<<<RH, TB, 0, stream>>>(hrW,                 phr0, HDIM, HDIM);
  repack_kernel<<<RH, TB, 0, stream>>>(hrW + HDIM * HDIM,   phr1, HDIM, HDIM);
  repack_kernel<<<RH, TB, 0, stream>>>(hhW,                 phh0, HDIM, HDIM);
  repack_kernel<<<RH, TB, 0, stream>>>(hhW + HDIM * HDIM,   phh1, HDIM, HDIM);

  const int ROWB = cdiv(N, 16);  // 3125 row tiles

  for (int t = 0; t < T_STEPS; ++t) {
    xm_kernel<<<cdiv(N, TB), TB, 0, stream>>>(Xseq, uprev, vprev, xm, N, t);

    zero_kernel<<<cdiv((long)N * XPAD, TB), TB, 0, stream>>>(aggx, (long)N * XPAD);
    zero_kernel<<<cdiv((long)N * HDIM, TB), TB, 0, stream>>>(aggh, (long)N * HDIM);
    zero_kernel<<<cdiv((long)N * HDIM, TB), TB, 0, stream>>>(agghr, (long)N * HDIM);

    scatter_x_kernel<<<cdiv(E, TB), TB, 0, stream>>>(src, dst, nrm, xm, aggx, E);
    scatter_h_kernel<<<cdiv((long)E * 16, TB), TB, 0, stream>>>(src, dst, nrm, h, aggh, E);

    gates_kernel<<<ROWB, TB, 0, stream>>>(xm, aggx, h, aggh,
                                          pxz0, pxz1, pxr0, pxr1,
                                          phz0, phz1, phr0, phr1,
                                          xzb, xrb, hzb, hrb,
                                          zbuf, hrbuf, N);

    scatter_h_kernel<<<cdiv((long)E * 16, TB), TB, 0, stream>>>(src, dst, nrm, hrbuf, agghr, E);

    cand_kernel<<<ROWB, TB, 0, stream>>>(xm, aggx, hrbuf, agghr,
                                         pxh0, pxh1, phh0, phh1,
                                         xhb, hhb, zbuf, h, N);

    head_kernel<<<cdiv(N, TB), TB, 0, stream>>>(h, headW, headb, Xseq, xm,
                                                out, uprev, vprev, N, t);
  }
  (void)n_in; (void)out_size; (void)ws_size;
}